// SecureOptimizedBlockReLU_17265768530070
// MI455X (gfx1250) — compile-verified
//
#include <hip/hip_runtime.h>
#include <hip/hip_bf16.h>
#include <stdint.h>

// ---------------------------------------------------------------------------
// BlockReLU for (N,16,224,224) fp32 — single fused kernel.
//   ch 0-1 : identity                     (phase 0)
//   ch 2-3 : elementwise sign mask (1x1)  (phase 0)
//   ch 4-7 : 2x2 block-sum sign mask      (phase 1)
//   ch 8-11: 4x4 block-sum sign mask      (phase 2)
//   ch12-15: 3x3 block-sum sign mask      (phase 3, padded 224->225, async LDS)
// Memory bound (~206 MB, AI ~0.25 flop/B) -> no WMMA; one dispatch, all
// phases partitioned over blockIdx.x so the whole chip streams concurrently.
// ---------------------------------------------------------------------------

#define CH   16
#define HH   224
#define WW   224
#define HWSZ (HH * WW)        // 50176 floats per channel image
#define HW4  (HWSZ / 4)       // 12544 float4 per channel image

typedef int v4i __attribute__((ext_vector_type(4)));

// ---- gfx1250 async global->LDS helpers (builtin if present, else asm) -----
__device__ __forceinline__ void async_g2l_b128(const float* gsrc, float* ldst) {
#if __has_builtin(__builtin_amdgcn_global_load_async_to_lds_b128)
  __builtin_amdgcn_global_load_async_to_lds_b128(
      (__attribute__((address_space(1))) v4i*)gsrc,
      (__attribute__((address_space(3))) v4i*)ldst, 0, 0);
#else
  unsigned lds_off =
      (unsigned)(uintptr_t)(__attribute__((address_space(3))) void*)ldst;
  asm volatile("global_load_async_to_lds_b128 %0, %1, off"
               :: "v"(lds_off), "v"(gsrc) : "memory");
#endif
}

__device__ __forceinline__ void wait_async0() {
#if __has_builtin(__builtin_amdgcn_s_wait_asynccnt)
  __builtin_amdgcn_s_wait_asynccnt(0);
#else
  asm volatile("s_wait_asynccnt 0" ::: "memory");
#endif
}

// Block ranges per N (blocks of 256 threads):
//   phase0: N*196  (ch 0-3, 49 blocks per channel image)
//   phase1: N*98   (ch 4-7)
//   phase2: N*49   (ch 8-11)
//   phase3: N*76   (ch 12-15, 19 groups of 4 block-rows per channel image)
__global__ __launch_bounds__(256) void k_blockrelu(const float* __restrict__ in,
                                                   float* __restrict__ out,
                                                   int N) {
  __shared__ __align__(16) float xs[4][3][WW];     // phase3 raw strips
  __shared__ __align__(16) float colsum[4][228];   // phase3 col sums + pad

  const int T1 = N * 196;
  const int T2 = T1 + N * 98;
  const int T3 = T2 + N * 49;

  int b = blockIdx.x;
  int t = threadIdx.x;

  if (b < T1) {
    // ---- phase 0: ch 0-3, identity / elementwise mask; float4 per thread --
    int img = b / 49;                 // n*4 + c   (uniform, SALU)
    int pos = (b - img * 49) * 256 + t;
    int n   = img >> 2;
    int c   = img & 3;
    int off = (n * CH + c) * HW4 + pos;
    float4 x = ((const float4*)in)[off];
    if (c >= 2) {
      x.x *= (x.x >= 0.0f) ? 1.0f : 0.0f;
      x.y *= (x.y >= 0.0f) ? 1.0f : 0.0f;
      x.z *= (x.z >= 0.0f) ? 1.0f : 0.0f;
      x.w *= (x.w >= 0.0f) ? 1.0f : 0.0f;
    }
    ((float4*)out)[off] = x;

  } else if (b < T2) {
    // ---- phase 1: ch 4-7, 2x2 blocks; thread owns 2 rows x 4 cols ---------
    int idx  = (b - T1) * 256 + t;
    int col4 = idx % 56;
    int br   = (idx / 56) % 112;
    int cg   = (idx / (56 * 112)) & 3;
    int n    = idx / (56 * 112 * 4);
    int base = ((n * CH + 4 + cg) * HH + br * 2) * WW + col4 * 4;

    float4 a = *(const float4*)(in + base);
    float4 c = *(const float4*)(in + base + WW);

    float s0 = (a.x + a.y) + (c.x + c.y);
    float s1 = (a.z + a.w) + (c.z + c.w);
    float m0 = (s0 >= 0.0f) ? 1.0f : 0.0f;
    float m1 = (s1 >= 0.0f) ? 1.0f : 0.0f;

    a.x *= m0; a.y *= m0; a.z *= m1; a.w *= m1;
    c.x *= m0; c.y *= m0; c.z *= m1; c.w *= m1;

    *(float4*)(out + base)      = a;
    *(float4*)(out + base + WW) = c;

  } else if (b < T3) {
    // ---- phase 2: ch 8-11, 4x4 blocks; thread owns exactly one block ------
    int idx  = (b - T2) * 256 + t;
    int col4 = idx % 56;
    int br   = (idx / 56) % 56;
    int cg   = (idx / (56 * 56)) & 3;
    int n    = idx / (56 * 56 * 4);
    int base = ((n * CH + 8 + cg) * HH + br * 4) * WW + col4 * 4;

    float4 r0 = *(const float4*)(in + base);
    float4 r1 = *(const float4*)(in + base + WW);
    float4 r2 = *(const float4*)(in + base + 2 * WW);
    float4 r3 = *(const float4*)(in + base + 3 * WW);

    float s = ((r0.x + r0.y) + (r0.z + r0.w)) + ((r1.x + r1.y) + (r1.z + r1.w)) +
              ((r2.x + r2.y) + (r2.z + r2.w)) + ((r3.x + r3.y) + (r3.z + r3.w));
    float m = (s >= 0.0f) ? 1.0f : 0.0f;

    r0.x *= m; r0.y *= m; r0.z *= m; r0.w *= m;
    r1.x *= m; r1.y *= m; r1.z *= m; r1.w *= m;
    r2.x *= m; r2.y *= m; r2.z *= m; r2.w *= m;
    r3.x *= m; r3.y *= m; r3.z *= m; r3.w *= m;

    *(float4*)(out + base)          = r0;
    *(float4*)(out + base + WW)     = r1;
    *(float4*)(out + base + 2 * WW) = r2;
    *(float4*)(out + base + 3 * WW) = r3;

  } else {
    // ---- phase 3: ch 12-15, 3x3 blocks (padded 224->225 both dims) --------
    // 256 threads = 4 sub-strips x 64 lanes; lanes 0-55 active (56 float4
    // columns), lanes 56-63 idle but participate in barriers. Each sub-strip
    // = one block-row (3 image rows) staged in LDS via async b128 loads.
    int g    = b - T3;
    int grp  = g % 19;
    int cg   = (g / 19) & 3;
    int n    = g / (19 * 4);

    int sub  = t >> 6;
    int lane = t & 63;
    bool lactive = lane < 56;

    int br      = grp * 4 + sub;                // block row, valid if < 75
    bool valid  = (br < 75) && lactive;
    int imgBase = (n * CH + 12 + cg) * HWSZ;
    int r0      = br * 3;
    int colb    = lane * 4;

    if (valid) {
#pragma unroll
      for (int i = 0; i < 3; ++i) {
        int row = r0 + i;
        float* ldst = &xs[sub][i][colb];
        if (row < HH) {
          async_g2l_b128(in + imgBase + row * WW + colb, ldst);
        } else {                                // zero pad row 224 (br==74)
          *(float4*)ldst = make_float4(0.f, 0.f, 0.f, 0.f);
        }
      }
    }
    if (lane == 56) {                           // zero pad column 224
      colsum[sub][224] = 0.0f; colsum[sub][225] = 0.0f;
      colsum[sub][226] = 0.0f; colsum[sub][227] = 0.0f;
    }
    wait_async0();
    __syncthreads();

    float4 x0 = make_float4(0.f, 0.f, 0.f, 0.f);
    float4 x1 = x0, x2 = x0;
    if (lactive) {
      x0 = *(const float4*)&xs[sub][0][colb];
      x1 = *(const float4*)&xs[sub][1][colb];
      x2 = *(const float4*)&xs[sub][2][colb];
      float4 cs;
      cs.x = x0.x + x1.x + x2.x;
      cs.y = x0.y + x1.y + x2.y;
      cs.z = x0.z + x1.z + x2.z;
      cs.w = x0.w + x1.w + x2.w;
      *(float4*)&colsum[sub][colb] = cs;
    }
    __syncthreads();

    if (valid) {
      float m[4];
#pragma unroll
      for (int j = 0; j < 4; ++j) {
        int b3   = ((colb + j) / 3) * 3;
        float bs = colsum[sub][b3] + colsum[sub][b3 + 1] + colsum[sub][b3 + 2];
        m[j]     = (bs >= 0.0f) ? 1.0f : 0.0f;
      }
#pragma unroll
      for (int i = 0; i < 3; ++i) {
        int row = r0 + i;
        if (row < HH) {
          float4 x = (i == 0) ? x0 : (i == 1) ? x1 : x2;
          x.x *= m[0]; x.y *= m[1]; x.z *= m[2]; x.w *= m[3];
          *(float4*)(out + imgBase + row * WW + colb) = x;
        }
      }
    }
  }
}

// ---------------------------------------------------------------------------
extern "C" void kernel_launch(void* const* d_in, const int* in_sizes, int n_in,
                              void* d_out, int out_size, void* d_ws, size_t ws_size,
                              hipStream_t stream) {
  (void)n_in; (void)d_ws; (void)ws_size; (void)out_size;
  const float* in = (const float*)d_in[0];
  float* out      = (float*)d_out;
  int N = in_sizes[0] / (CH * HWSZ);            // 32 for the reference shape

  // One fused dispatch: N * (196 + 98 + 49 + 76) workgroups of 256 threads.
  k_blockrelu<<<dim3(N * 419), dim3(256), 0, stream>>>(in, out, N);
}